// Header_18485539242052
// MI455X (gfx1250) — compile-verified
//
#include <hip/hip_runtime.h>
#include <hip/hip_bf16.h>

#define NANCH   25200
#define BIMG    16
#define NCLS    80
#define TOPK    128
#define MAXOUT  100
#define SCTHR   0.25f
#define IOUTHR  0.45f

typedef __attribute__((ext_vector_type(2))) float v2f;
typedef __attribute__((ext_vector_type(8))) float v8f;
typedef unsigned long long u64;

__constant__ float c_anch[9][2] = {
    {10.f,13.f},{16.f,30.f},{33.f,23.f},
    {30.f,61.f},{62.f,45.f},{59.f,119.f},
    {116.f,90.f},{156.f,198.f},{373.f,326.f}};

__device__ __forceinline__ unsigned ordkey(float f) {
    unsigned u = __float_as_uint(f);
    return (u & 0x80000000u) ? ~u : (u | 0x80000000u);
}
__device__ __forceinline__ float ordinv(unsigned k) {
    unsigned u = (k & 0x80000000u) ? (k & 0x7FFFFFFFu) : ~k;
    return __uint_as_float(u);
}
__device__ __forceinline__ float sigmoidf(float x) {
    return 1.0f / (1.0f + expf(-x));
}

// ---------------- Kernel 1: decode + class-major scores ----------------
__global__ void __launch_bounds__(256)
decode_kernel(const float* __restrict__ p3, const float* __restrict__ p4,
              const float* __restrict__ p5, float* __restrict__ boxes,
              float* __restrict__ scoresT) {
    int s = blockIdx.x * blockDim.x + threadIdx.x;
    if (s >= BIMG * NANCH) return;
    int b = s / NANCH, r = s % NANCH;
    const float* src; int H, W, lvl, rl; float stride;
    if (r < 19200)      { src = p3; H = 80; W = 80; stride =  8.f; lvl = 0; rl = r; }
    else if (r < 24000) { src = p4; H = 40; W = 40; stride = 16.f; lvl = 1; rl = r - 19200; }
    else                { src = p5; H = 20; W = 20; stride = 32.f; lvl = 2; rl = r - 24000; }
    int cell = rl / 3, a = rl % 3;
    int h = cell / W, w = cell % W;
    const float* p = src + (((size_t)(b * H + h)) * W + w) * 255 + a * 85;

    float sx = sigmoidf(p[0]), sy = sigmoidf(p[1]);
    float aw = c_anch[lvl * 3 + a][0], ah = c_anch[lvl * 3 + a][1];
    float cx = (sx + (float)w) * stride, cy = (sy + (float)h) * stride;
    float bw = expf(p[2]) * aw, bh = expf(p[3]) * ah;
    size_t bo = ((size_t)b * NANCH + r) * 4;
    boxes[bo + 0] = cx - bw * 0.5f;
    boxes[bo + 1] = cy - bh * 0.5f;
    boxes[bo + 2] = cx + bw * 0.5f;
    boxes[bo + 3] = cy + bh * 0.5f;

    float so = sigmoidf(p[4]);
    size_t base = ((size_t)b * NCLS) * NANCH + r;
#pragma unroll 4
    for (int c = 0; c < NCLS; ++c) {
        float sc = sigmoidf(p[5 + c]);
        scoresT[base + (size_t)c * NANCH] = so * sc;   // coalesced across wave per class
    }
}

// ---------------- Hybrid register/LDS bitonic (wave32-aware) ----------------
// Element id == thread id (0..255). Distances j<32 stay inside a wave32 and use
// lane shuffles (no barrier); j in {32,64,128} round-trips through LDS.
__device__ __forceinline__ u64 shflx64(u64 v, int m) {
    unsigned lo = __shfl_xor((unsigned)v, m, 32);
    unsigned hi = __shfl_xor((unsigned)(v >> 32), m, 32);
    return ((u64)hi << 32) | lo;
}
__device__ __forceinline__ u64 xchg(u64 v, u64* sh, int t, int j) {
    u64 w;
    if (j >= 32) {
        sh[t] = v; __syncthreads();
        w = sh[t ^ j]; __syncthreads();
    } else {
        w = shflx64(v, j);
    }
    return w;
}
__device__ __forceinline__ u64 sort256_asc(u64 v, u64* sh, int t) {
#pragma unroll
    for (int k = 2; k <= 256; k <<= 1) {
#pragma unroll
        for (int j = k >> 1; j > 0; j >>= 1) {
            u64 w = xchg(v, sh, t, j);
            bool up = ((t & k) == 0);
            bool lower = ((t & j) == 0);
            u64 mn = (v < w) ? v : w;
            u64 mx = (v < w) ? w : v;
            v = (up == lower) ? mn : mx;
        }
    }
    return v;
}
__device__ __forceinline__ u64 merge256_desc(u64 v, u64* sh, int t) {
#pragma unroll
    for (int j = 128; j > 0; j >>= 1) {
        u64 w = xchg(v, sh, t, j);
        bool lower = ((t & j) == 0);
        u64 mn = (v < w) ? v : w;
        u64 mx = (v < w) ? w : v;
        v = lower ? mx : mn;
    }
    return v;
}

// ---------------- Kernel 2: top-128 per (image,class) ----------------
__global__ void __launch_bounds__(256)
topk_kernel(const float* __restrict__ scoresT, float* __restrict__ topv,
            int* __restrict__ topi) {
    __shared__ u64 sh[256];
    int t = threadIdx.x;
    int bc = blockIdx.x;                     // b*80 + c
    const float* row = scoresT + (size_t)bc * NANCH;
    u64 vb = 0ULL;                           // buffer element (descending across block)
    float s_next = row[t];                   // software-pipelined chunk load
    for (int c0 = 0; c0 < NANCH; c0 += 256) {
        int pos = c0 + t;
        float s_cur = s_next;
        int nxt = pos + 256;
        if (nxt < NANCH) s_next = row[nxt];  // prefetch next chunk before the sort
        u64 vc = 0ULL;
        if (pos < NANCH) {
            float sc = (s_cur > SCTHR) ? s_cur : -1.0f;
            vc = ((u64)ordkey(sc) << 32) | (u64)(0xFFFFFFFFu - (unsigned)pos);
        }
        vc = sort256_asc(vc, sh, t);         // chunk ascending
        vb = (vb > vc) ? vb : vc;            // valley merge in registers
        vb = merge256_desc(vb, sh, t);       // buffer back to descending
    }
    if (t < TOPK) {
        int idx = (int)(0xFFFFFFFFu - (unsigned)(vb & 0xFFFFFFFFu));
        if (idx < 0) idx = 0;
        if (idx >= NANCH) idx = NANCH - 1;
        topv[bc * TOPK + t] = ordinv((unsigned)(vb >> 32));
        topi[bc * TOPK + t] = idx;
    }
}

// ---------------- Kernel 3: NMS per (image,class); WMMA for area sums ----------------
__global__ void __launch_bounds__(256)
nms_kernel(const float* __restrict__ boxes, const float* __restrict__ topv,
           const int* __restrict__ topi, float* __restrict__ cb,
           float* __restrict__ cs) {
    __shared__ unsigned char sup[TOPK * TOPK];           // 16 KB: iou > THR mask
    __shared__ float lx1[TOPK], ly1[TOPK], lx2[TOPK], ly2[TOPK], lar[TOPK], lsc[TOPK];
    __shared__ int keep[TOPK], pos[TOPK];
    int t = threadIdx.x;
    int bc = blockIdx.x;
    int b = bc / NCLS;

    if (t < TOPK) {
        int idx = topi[bc * TOPK + t];
        const float* bp = boxes + ((size_t)b * NANCH + idx) * 4;
        float a0 = bp[0], a1 = bp[1], a2 = bp[2], a3 = bp[3];
        lx1[t] = a0; ly1[t] = a1; lx2[t] = a2; ly2[t] = a3;
        lar[t] = (a2 - a0) * (a3 - a1);
        lsc[t] = topv[bc * TOPK + t];
    }
    __syncthreads();

    // 128x128 (area_i + area_j) via V_WMMA_F32_16X16X4_F32 rank-2 outer product:
    // A[m] = [area_m, 1, 0, 0]; B[:,n] = [1, area_n, 0, 0]^T => D = area_m + area_n (exact)
    int lane = t & 31, wave = t >> 5;
    bool hi = lane >= 16;
    int l16 = lane & 15;
    for (int tile = wave; tile < 64; tile += 8) {        // 8x8 grid of 16x16 tiles
        int tr = tile >> 3, tc = tile & 7;
        v2f Af, Bf;
        Af.x = hi ? 0.f : lar[tr * 16 + l16];            // K=0 row (lanes 0-15)
        Af.y = hi ? 0.f : 1.0f;                          // K=1
        Bf.x = hi ? 0.f : 1.0f;                          // B row K=0
        Bf.y = hi ? 0.f : lar[tc * 16 + l16];            // B row K=1
        v8f Cf = {};
        v8f Df = __builtin_amdgcn_wmma_f32_16x16x4_f32(
            false, Af, false, Bf, (short)0, Cf, false, false);
        int gn = tc * 16 + l16;
        float bx1 = lx1[gn], by1 = ly1[gn], bx2 = lx2[gn], by2 = ly2[gn];
#pragma unroll
        for (int r = 0; r < 8; ++r) {
            int gm = tr * 16 + r + (hi ? 8 : 0);
            float xx1 = fmaxf(lx1[gm], bx1);
            float yy1 = fmaxf(ly1[gm], by1);
            float xx2 = fminf(lx2[gm], bx2);
            float yy2 = fminf(ly2[gm], by2);
            float inter = fmaxf(xx2 - xx1, 0.f) * fmaxf(yy2 - yy1, 0.f);
            float iou = inter / (Df[r] - inter + 1e-9f);
            sup[gm * TOPK + gn] = (iou > IOUTHR) ? 1 : 0;
        }
    }
    __syncthreads();

    // Greedy suppression entirely inside wave 0: 128-bit keep state lives in
    // registers (4 slots/lane), keep[i] broadcast via __shfl, zero barriers.
    if (wave == 0) {
        unsigned kbit[4] = {1u, 1u, 1u, 1u};             // lane owns j = slot*32+lane
#pragma unroll
        for (int slot = 0; slot < 4; ++slot) {
            for (int li = 0; li < 32; ++li) {
                int i = slot * 32 + li;
                unsigned ki = __shfl(kbit[slot], li, 32);
                float si = lsc[i];
                if (ki && si > 0.f) {
#pragma unroll
                    for (int s2 = 0; s2 < 4; ++s2) {
                        int j = s2 * 32 + lane;
                        if (j > i && sup[i * TOPK + j]) kbit[s2] = 0u;
                    }
                }
            }
        }
#pragma unroll
        for (int slot = 0; slot < 4; ++slot) {
            int j = slot * 32 + lane;
            keep[j] = (kbit[slot] != 0u) && (lsc[j] > 0.f);
        }
    }
    __syncthreads();

    if (t == 0) {
        int c = 0;
        for (int i = 0; i < TOPK; ++i) { pos[i] = c; c += keep[i]; }
    }
    __syncthreads();

    float* outs = cs + (size_t)bc * MAXOUT;
    float* outb = cb + (size_t)bc * MAXOUT * 4;
    if (t < MAXOUT) {
        outs[t] = -1.0f;
        outb[t * 4 + 0] = 0.f; outb[t * 4 + 1] = 0.f;
        outb[t * 4 + 2] = 0.f; outb[t * 4 + 3] = 0.f;
    }
    __syncthreads();
    if (t < TOPK && keep[t]) {
        int p = pos[t];
        if (p < MAXOUT) {
            outs[p] = lsc[t];
            outb[p * 4 + 0] = lx1[t]; outb[p * 4 + 1] = ly1[t];
            outb[p * 4 + 2] = lx2[t]; outb[p * 4 + 3] = ly2[t];
        }
    }
}

// ---------------- Kernel 4: per-image merge (top-100 of 8000) ----------------
__global__ void __launch_bounds__(256)
merge_kernel(const float* __restrict__ cb, const float* __restrict__ cs,
             float* __restrict__ out) {
    __shared__ u64 sh[256];
    __shared__ int cnt;
    int t = threadIdx.x, b = blockIdx.x;
    const int N = NCLS * MAXOUT;                     // 8000
    const float* rs = cs + (size_t)b * N;
    if (t == 0) cnt = 0;
    u64 vb = 0ULL;
    float s_next = rs[t];
    for (int c0 = 0; c0 < N; c0 += 256) {
        int p = c0 + t;
        float s_cur = s_next;
        int nxt = p + 256;
        if (nxt < N) s_next = rs[nxt];
        u64 vc = 0ULL;
        if (p < N) {
            vc = ((u64)ordkey(s_cur) << 32) | (u64)(0xFFFFFFFFu - (unsigned)p);
        }
        vc = sort256_asc(vc, sh, t);
        vb = (vb > vc) ? vb : vc;
        vb = merge256_desc(vb, sh, t);
    }
    float* ob = out;                                  // (16,100,4)
    float* os = out + BIMG * MAXOUT * 4;              // (16,100)
    float* oc = os + BIMG * MAXOUT;                   // (16,100)
    float* on = oc + BIMG * MAXOUT;                   // (16,)
    if (t < MAXOUT) {
        float sc = ordinv((unsigned)(vb >> 32));
        int fi = (int)(0xFFFFFFFFu - (unsigned)(vb & 0xFFFFFFFFu));
        if (fi < 0) fi = 0;
        if (fi >= N) fi = N - 1;
        bool valid = sc > 0.f;
        const float* bb = cb + ((size_t)b * N + fi) * 4;
        size_t o = (size_t)b * MAXOUT + t;
        ob[o * 4 + 0] = valid ? bb[0] : 0.f;
        ob[o * 4 + 1] = valid ? bb[1] : 0.f;
        ob[o * 4 + 2] = valid ? bb[2] : 0.f;
        ob[o * 4 + 3] = valid ? bb[3] : 0.f;
        os[o] = valid ? sc : 0.f;
        oc[o] = valid ? (float)(fi / MAXOUT) : 0.f;
        if (valid) atomicAdd(&cnt, 1);
    }
    __syncthreads();
    if (t == 0) on[b] = (float)cnt;
}

// ---------------- Host launcher ----------------
extern "C" void kernel_launch(void* const* d_in, const int* in_sizes, int n_in,
                              void* d_out, int out_size, void* d_ws, size_t ws_size,
                              hipStream_t stream) {
    const float* p3 = (const float*)d_in[0];
    const float* p4 = (const float*)d_in[1];
    const float* p5 = (const float*)d_in[2];
    char* ws = (char*)d_ws;
    // Workspace layout (bytes)
    float* boxes   = (float*)(ws + 0);                    //  25,804,800 B
    float* scoresT = (float*)(ws + 25804800);             // 129,024,000 B
    float* topv    = (float*)(ws + 154828800);            //     655,360 B
    int*   topi    = (int*)  (ws + 155484160);            //     655,360 B
    float* csb     = (float*)(ws + 156139520);            //     512,000 B
    float* cbb     = (float*)(ws + 156651520);            //   2,048,000 B

    int total = BIMG * NANCH;                             // 403,200
    decode_kernel<<<(total + 255) / 256, 256, 0, stream>>>(p3, p4, p5, boxes, scoresT);
    topk_kernel<<<BIMG * NCLS, 256, 0, stream>>>(scoresT, topv, topi);
    nms_kernel<<<BIMG * NCLS, 256, 0, stream>>>(boxes, topv, topi, cbb, csb);
    merge_kernel<<<BIMG, 256, 0, stream>>>(cbb, csb, (float*)d_out);
}